// SinkhornDistance_7954279432503
// MI455X (gfx1250) — compile-verified
//
#include <hip/hip_runtime.h>

typedef __attribute__((ext_vector_type(2))) float v2f;
typedef __attribute__((ext_vector_type(8))) float v8f;

#define NROW 4096
#define KDIM 256
#define EPSV 0.1f
#define INV_EPS 10.0f
#define NEG_BIG (-3.4e38f)

// ---------------- init v = 0 (u is overwritten before first read) ------------
__global__ void sink_init(float* __restrict__ u, float* __restrict__ v) {
    int g = blockIdx.x * blockDim.x + threadIdx.x;
    if (g < NROW) { u[g] = 0.0f; v[g] = 0.0f; }
}

// ---------------- row squared norms: x2[i] = sum_k x[i][k]^2 -----------------
__global__ __launch_bounds__(256)
void sink_rowsq(const float* __restrict__ x, const float* __restrict__ y,
                float* __restrict__ x2, float* __restrict__ y2) {
    __shared__ float sd[256];
    const int row = blockIdx.x;
    const float* p = (blockIdx.y == 0 ? x : y) + (size_t)row * KDIM;
    float a = p[threadIdx.x];           // KDIM == 256 == blockDim
    sd[threadIdx.x] = a * a;
    __syncthreads();
    for (int off = 128; off > 0; off >>= 1) {
        if (threadIdx.x < off) sd[threadIdx.x] += sd[threadIdx.x + off];
        __syncthreads();
    }
    if (threadIdx.x == 0) (blockIdx.y == 0 ? x2 : y2)[row] = sd[0];
}

// ---------------- WMMA f32 GEMM: C = x2 + y2 - 2 * x @ y^T -------------------
// One wave computes a 16(M) x 64(N) strip. A fragment reused across 4 N-tiles.
__global__ __launch_bounds__(32)
void sink_cost_wmma(const float* __restrict__ x, const float* __restrict__ y,
                    const float* __restrict__ x2, const float* __restrict__ y2,
                    float* __restrict__ C) {
    const int lane  = threadIdx.x;          // 0..31
    const int r     = lane & 15;
    const int h     = lane >> 4;            // 0 or 1 -> K half
    const int mBase = blockIdx.y * 16;      // 256 M-tiles
    const int nBase = blockIdx.x * 64;      // 64 N-strips

    v8f acc0 = {}, acc1 = {}, acc2 = {}, acc3 = {};

    const float* xrow  = x + (size_t)(mBase + r) * KDIM;   // A: row m = mBase+r
    const float* yrow0 = y + (size_t)(nBase + r) * KDIM;   // B tile 0: n = nBase+r
    const float* yrow1 = yrow0 + 16 * KDIM;
    const float* yrow2 = yrow0 + 32 * KDIM;
    const float* yrow3 = yrow0 + 48 * KDIM;

    for (int k = 0; k < KDIM; k += 4) {
        const int kb = k + h * 2;           // lane's K pair within the 4-step
        v2f a  = *(const v2f*)(xrow  + kb);
        v2f b0 = *(const v2f*)(yrow0 + kb);
        v2f b1 = *(const v2f*)(yrow1 + kb);
        v2f b2 = *(const v2f*)(yrow2 + kb);
        v2f b3 = *(const v2f*)(yrow3 + kb);
        acc0 = __builtin_amdgcn_wmma_f32_16x16x4_f32(false, a, false, b0, (short)0, acc0, false, false);
        acc1 = __builtin_amdgcn_wmma_f32_16x16x4_f32(false, a, false, b1, (short)0, acc1, false, false);
        acc2 = __builtin_amdgcn_wmma_f32_16x16x4_f32(false, a, false, b2, (short)0, acc2, false, false);
        acc3 = __builtin_amdgcn_wmma_f32_16x16x4_f32(false, a, false, b3, (short)0, acc3, false, false);
    }

    // C/D layout: VGPR rr -> M = rr + 8*h (lanes 0-15: M=rr; 16-31: M=rr+8), N = lane&15
    const int n0 = nBase + r;
    const float ys0 = y2[n0], ys1 = y2[n0 + 16], ys2 = y2[n0 + 32], ys3 = y2[n0 + 48];
#pragma unroll
    for (int rr = 0; rr < 8; ++rr) {
        const int m  = mBase + rr + 8 * h;
        const float xs = x2[m];
        float* Crow = C + (size_t)m * NROW;
        Crow[n0]      = xs + ys0 - 2.0f * acc0[rr];
        Crow[n0 + 16] = xs + ys1 - 2.0f * acc1[rr];
        Crow[n0 + 32] = xs + ys2 - 2.0f * acc2[rr];
        Crow[n0 + 48] = xs + ys3 - 2.0f * acc3[rr];
    }
}

// ---------------- online (max, sum) logsumexp helpers ------------------------
__device__ __forceinline__ void lse_push(float t, float& m, float& s) {
    if (t > m) { s *= __expf(m - t); m = t; }
    s += __expf(t - m);
}
__device__ __forceinline__ void lse_merge(float& m1, float& s1, float m2, float s2) {
    float mo = fmaxf(m1, m2);
    s1 = s1 * __expf(m1 - mo) + s2 * __expf(m2 - mo);
    m1 = mo;
}

// ---------------- u[i] = -eps * LSE_j((v[j] - C[i][j]) / eps) ----------------
__global__ __launch_bounds__(256)
void sink_row_lse(const float* __restrict__ C, const float* __restrict__ v,
                  float* __restrict__ u) {
    __shared__ float sm[256], ss[256];
    const int row = blockIdx.x, tid = threadIdx.x;
    const float* Crow = C + (size_t)row * NROW;
    float m = NEG_BIG, s = 0.0f;
    for (int j = tid; j < NROW; j += 256)
        lse_push((v[j] - Crow[j]) * INV_EPS, m, s);
    sm[tid] = m; ss[tid] = s;
    __syncthreads();
    for (int off = 128; off > 0; off >>= 1) {
        if (tid < off) {
            float m1 = sm[tid], s1 = ss[tid];
            lse_merge(m1, s1, sm[tid + off], ss[tid + off]);
            sm[tid] = m1; ss[tid] = s1;
        }
        __syncthreads();
    }
    if (tid == 0) u[row] = -EPSV * (sm[0] + __logf(ss[0]));
}

// ---------------- column LSE, phase A: per-chunk partials --------------------
// grid (16, 16): blockIdx.x*256+tid = column, blockIdx.y = 256-row chunk
__global__ __launch_bounds__(256)
void sink_col_lse_partial(const float* __restrict__ C, const float* __restrict__ u,
                          float* __restrict__ pm, float* __restrict__ ps) {
    const int col = blockIdx.x * 256 + threadIdx.x;
    const int i0  = blockIdx.y * 256;
    float m = NEG_BIG, s = 0.0f;
    for (int i = i0; i < i0 + 256; ++i)
        lse_push((u[i] - C[(size_t)i * NROW + col]) * INV_EPS, m, s);
    pm[blockIdx.y * NROW + col] = m;
    ps[blockIdx.y * NROW + col] = s;
}

// ---------------- column LSE, phase B: fixed-order combine -------------------
__global__ __launch_bounds__(256)
void sink_col_lse_combine(const float* __restrict__ pm, const float* __restrict__ ps,
                          float* __restrict__ v) {
    const int col = blockIdx.x * 256 + threadIdx.x;
    float m = pm[col], s = ps[col];
    for (int c = 1; c < 16; ++c)
        lse_merge(m, s, pm[c * NROW + col], ps[c * NROW + col]);
    v[col] = -EPSV * (m + __logf(s));
}

// ---------------- cost = sum(exp((u+v-C)/eps) * C) / N, stage 1 --------------
__global__ __launch_bounds__(256)
void sink_cost_partial(const float* __restrict__ C, const float* __restrict__ u,
                       const float* __restrict__ v, float* __restrict__ part) {
    __shared__ float sd[256];
    const int tid = threadIdx.x;
    float acc = 0.0f;
    const size_t total = (size_t)NROW * NROW;
    for (size_t k = (size_t)blockIdx.x * 256 + tid; k < total; k += (size_t)4096 * 256) {
        const int i = (int)(k >> 12);
        const int j = (int)(k & 4095);
        const float c = C[k];
        acc += __expf((u[i] + v[j] - c) * INV_EPS) * c;
    }
    sd[tid] = acc;
    __syncthreads();
    for (int off = 128; off > 0; off >>= 1) {
        if (tid < off) sd[tid] += sd[tid + off];
        __syncthreads();
    }
    if (tid == 0) part[blockIdx.x] = sd[0];
}

// ---------------- cost, stage 2 (single block, deterministic tree) -----------
__global__ __launch_bounds__(256)
void sink_cost_final(const float* __restrict__ part, float* __restrict__ out) {
    __shared__ float sd[256];
    const int tid = threadIdx.x;
    float acc = 0.0f;
    for (int c = 0; c < 16; ++c) acc += part[c * 256 + tid];
    sd[tid] = acc;
    __syncthreads();
    for (int off = 128; off > 0; off >>= 1) {
        if (tid < off) sd[tid] += sd[tid + off];
        __syncthreads();
    }
    if (tid == 0) out[0] = sd[0] / (float)NROW;
}

extern "C" void kernel_launch(void* const* d_in, const int* in_sizes, int n_in,
                              void* d_out, int out_size, void* d_ws, size_t ws_size,
                              hipStream_t stream) {
    const float* x = (const float*)d_in[0];
    const float* y = (const float*)d_in[1];
    float* out = (float*)d_out;

    // workspace layout (floats)
    float* ws   = (float*)d_ws;
    float* Cm   = ws;                                    // 4096*4096
    float* x2   = Cm + (size_t)NROW * NROW;              // 4096
    float* y2   = x2 + NROW;                             // 4096
    float* u    = y2 + NROW;                             // 4096
    float* v    = u + NROW;                              // 4096
    float* pm   = v + NROW;                              // 16*4096
    float* ps   = pm + 16 * NROW;                        // 16*4096
    float* part = ps + 16 * NROW;                        // 4096

    // 1) init duals
    sink_init<<<dim3(16), dim3(256), 0, stream>>>(u, v);
    // 2) row squared norms for x and y
    sink_rowsq<<<dim3(NROW, 2), dim3(256), 0, stream>>>(x, y, x2, y2);
    // 3) cost matrix via f32 WMMA
    sink_cost_wmma<<<dim3(64, 256), dim3(32), 0, stream>>>(x, y, x2, y2, Cm);
    // 4) 100 Sinkhorn iterations (u from v, then v from new u)
    for (int it = 0; it < 100; ++it) {
        sink_row_lse<<<dim3(NROW), dim3(256), 0, stream>>>(Cm, v, u);
        sink_col_lse_partial<<<dim3(16, 16), dim3(256), 0, stream>>>(Cm, u, pm, ps);
        sink_col_lse_combine<<<dim3(16), dim3(256), 0, stream>>>(pm, ps, v);
    }
    // 5) transport cost reduction (deterministic two-stage)
    sink_cost_partial<<<dim3(4096), dim3(256), 0, stream>>>(Cm, u, v, part);
    sink_cost_final<<<dim3(1), dim3(256), 0, stream>>>(part, out);
}